// TFM_53171695125157
// MI455X (gfx1250) — compile-verified
//
#include <hip/hip_runtime.h>
#include <stdint.h>

typedef float v2f  __attribute__((ext_vector_type(2)));
typedef float v8f  __attribute__((ext_vector_type(8)));
typedef unsigned int u32x4 __attribute__((ext_vector_type(4)));
typedef int   i32x4 __attribute__((ext_vector_type(4)));
typedef int   i32x8 __attribute__((ext_vector_type(8)));

#define NB   1024
#define HIS  50
#define KPT  5
#define NPOS 250   // HIS*KPT
#define DIM  256
#define SIG  100

__global__ __launch_bounds__(256) void fuse_dedup_kernel(
    const float* __restrict__ terms,     // [B, NPOS, DIM]
    const int*   __restrict__ term_ids,  // [B, NPOS]
    const int*   __restrict__ his_news,  // [B, HIS, SIG]
    float*       __restrict__ out)       // [B, NPOS, DIM]
{
  __shared__ int s_his[HIS * SIG];   // 20 KB, staged by TDM
  __shared__ int s_ids[256];
  __shared__ int s_first[256];
  __shared__ int s_seg[256];
  __shared__ int s_next[256];
  __shared__ int s_head[256];
  __shared__ int s_cnt[256];
  __shared__ int s_tail[256];
  __shared__ int s_multi[256];       // segment ids with >1 contributor
  __shared__ int s_useg[256];        // multi-slot of each union-list entry
  __shared__ int s_ulist[256];       // union list of contributor row indices
  __shared__ int s_uoff[256];        // union-list start offset per multi slot
  __shared__ int s_nseg, s_nmulti, s_utot;

  const int tid = threadIdx.x;
  const int b   = blockIdx.x;

  // ---------------- Phase 0: TDM async-tensor stage of his_news[b] ----------
  if (tid == 0) {
    const uint64_t ga = (uint64_t)(uintptr_t)(his_news + (size_t)b * (HIS * SIG));
    const uint32_t lds_off = (uint32_t)(uintptr_t)(&s_his[0]);
    // D# group 0: count=1 (valid), lds_addr, global_addr[56:0], type=2
    u32x4 g0;
    g0[0] = 1u;
    g0[1] = lds_off;
    g0[2] = (uint32_t)ga;
    g0[3] = ((uint32_t)(ga >> 32) & 0x01FFFFFFu) | 0x80000000u;
    // D# group 1: mask=0, data_size=4B, dims: 5000x1 tile of 5000x1 tensor
    i32x8 g1;
    g1[0] = 0x20000;                                // data_size = 2 (4 bytes)
    g1[1] = (int)(((HIS * SIG) & 0xFFFF) << 16);    // tensor_dim0 = 5000
    g1[2] = 0x00010000;                             // tensor_dim1 = 1
    g1[3] = (int)(((HIS * SIG) & 0xFFFF) << 16);    // tile_dim0   = 5000
    g1[4] = 1;                                      // tile_dim1=1, tile_dim2=0
    g1[5] = HIS * SIG;                              // tensor_dim0_stride lo
    g1[6] = (int)(((HIS * SIG) & 0xFFFF) << 16);    // stride0 hi=0 | stride1 lo
    g1[7] = 0;
    // D# groups 2/3: higher dims collapsed to 1
    i32x4 g2; g2[0] = 1; g2[1] = 1; g2[2] = HIS * SIG; g2[3] = 0;
    i32x4 g3; g3[0] = HIS * SIG; g3[1] = 0x00010000; g3[2] = 0; g3[3] = 0;
    // clang-23 / therock-10.0 lane: 6-arg form with an extra (unused) v8i slot
    i32x8 g4; g4[0]=0; g4[1]=0; g4[2]=0; g4[3]=0; g4[4]=0; g4[5]=0; g4[6]=0; g4[7]=0;
    __builtin_amdgcn_tensor_load_to_lds(g0, g1, g2, g3, g4, 0);
    __builtin_amdgcn_s_wait_tensorcnt(0);
  }
  __syncthreads();

  // ---------------- Phase 1: ids, first-occurrence, segments ----------------
  if (tid < NPOS) {
    const int pos = term_ids[(size_t)b * NPOS + tid];   // 0..SIG-1
    const int h   = tid / KPT;
    s_ids[tid]  = s_his[h * SIG + pos];
    s_next[tid] = -1;
  }
  __syncthreads();

  if (tid < NPOS) {
    const int my = s_ids[tid];
    int f = tid;
    for (int m = 0; m < tid; ++m) {
      if (s_ids[m] == my) { f = m; break; }
    }
    s_first[tid] = f;
  }
  __syncthreads();

  if (tid == 0) {
    int ns = 0;
    for (int n = 0; n < NPOS; ++n) {
      if (s_first[n] == n) {
        s_seg[n] = ns; s_head[ns] = n; s_cnt[ns] = 1; s_tail[ns] = n; ++ns;
      } else {
        const int sg = s_seg[s_first[n]];
        s_seg[n] = sg;
        s_cnt[sg] += 1;
        s_next[s_tail[sg]] = n;
        s_tail[sg] = n;
      }
    }
    s_nseg = ns;
    int nm = 0, u = 0;
    for (int r = 0; r < ns; ++r) {
      if (s_cnt[r] > 1) {
        s_multi[nm] = r;
        s_uoff[nm]  = u;
        for (int node = s_head[r]; node >= 0; node = s_next[node]) {
          s_ulist[u] = node; s_useg[u] = nm; ++u;
        }
        ++nm;
      }
    }
    s_nmulti = nm; s_utot = u;
  }
  __syncthreads();

  // ---------------- Phase 2a: streaming dedup-gather (hot path) -------------
  const int wave = tid >> 5;
  const int lane = tid & 31;
  const int nseg = s_nseg;
  const size_t base = (size_t)b * NPOS * DIM;

  for (int r = wave; r < NPOS; r += 8) {
    float4* dst = (float4*)(out + base + (size_t)r * DIM + lane * 8);
    if (r >= nseg) {
      const float4 z = make_float4(0.f, 0.f, 0.f, 0.f);
      dst[0] = z; dst[1] = z;
    } else if (s_cnt[r] == 1) {
      const int n0 = s_head[r];
      const float4* src = (const float4*)(terms + base + (size_t)n0 * DIM + lane * 8);
      dst[0] = src[0]; dst[1] = src[1];
    }
    // multi-contributor rows written by wave 0 below (disjoint rows)
  }

  // ---------------- Phase 2b: one-hot WMMA segment reduction (rare) ---------
  if (tid < 32) {            // wave 0, EXEC == all 32 lanes (WMMA requirement)
    const int nm   = s_nmulti;
    const int utot = s_utot;
    const int myM  = lane & 15;
    const int kh   = lane >> 4;        // lanes 0-15 vs 16-31

    for (int g = 0; g * 16 < nm; ++g) {
      const int mbase  = g * 16;
      const int kstart = s_uoff[mbase];
      const int kend   = (mbase + 16 < nm) ? s_uoff[mbase + 16] : utot;

      for (int t = 0; t < 16; ++t) {
        const int colbase = t * 16;
        v8f acc = {0.f, 0.f, 0.f, 0.f, 0.f, 0.f, 0.f, 0.f};

        for (int kc = kstart; kc < kend; kc += 4) {
          v2f a, bmat;
          // A 16x4 f32: lanes 0-15 -> {K0,K1}, lanes 16-31 -> {K2,K3}; M = lane&15
#pragma unroll
          for (int j = 0; j < 2; ++j) {
            const int kg = kc + kh * 2 + j;
            a[j] = (kg < kend && (s_useg[kg] - mbase) == myM) ? 1.0f : 0.0f;
          }
          // B 4x16 f32: VGPR j holds row K=j (lanes 0-15) / K=j+2 (lanes 16-31)
#pragma unroll
          for (int j = 0; j < 2; ++j) {
            const int kg = kc + j + kh * 2;
            bmat[j] = (kg < kend)
                        ? terms[base + (size_t)s_ulist[kg] * DIM + colbase + myM]
                        : 0.0f;
          }
          acc = __builtin_amdgcn_wmma_f32_16x16x4_f32(
              false, a, false, bmat, (short)0, acc, false, false);
        }

        // D 16x16 f32: VGPR j -> rows M=j (lanes 0-15) and M=j+8 (lanes 16-31)
#pragma unroll
        for (int j = 0; j < 8; ++j) {
          const int m  = j + kh * 8;
          const int mi = mbase + m;
          if (mi < nm) {
            out[base + (size_t)s_multi[mi] * DIM + colbase + myM] = acc[j];
          }
        }
      }
    }
  }
}

extern "C" void kernel_launch(void* const* d_in, const int* in_sizes, int n_in,
                              void* d_out, int out_size, void* d_ws, size_t ws_size,
                              hipStream_t stream) {
  (void)in_sizes; (void)n_in; (void)out_size; (void)d_ws; (void)ws_size;
  const float* terms    = (const float*)d_in[0];
  const int*   term_ids = (const int*)d_in[1];
  const int*   his_news = (const int*)d_in[2];
  float*       out      = (float*)d_out;
  fuse_dedup_kernel<<<dim3(NB), dim3(256), 0, stream>>>(terms, term_ids, his_news, out);
}